// MaxTimesPlusDilationLiftingP4_12987981103167
// MI455X (gfx1250) — compile-verified
//
#include <hip/hip_runtime.h>
#include <stdint.h>

// Problem constants (fixed by the reference)
#define B_  4
#define H_  64
#define W_  64
#define C_  16
#define F_  32
#define TW  8
#define TH  8
#define HALO_W (TW + 2)   // 10
#define HALO_H (TH + 2)   // 10
#define KVEC4  ((9 * C_ * F_) / 4)     // 1152 float4 per kernel tensor
#define XVEC4  ((HALO_H * HALO_W * C_) / 4)  // 400 float4 in x halo tile

// Issue one CDNA5 async global->LDS 128-bit copy (GVS mode: saddr + 32-bit voffset).
__device__ __forceinline__ void async_g2l_b128(unsigned lds_byte_addr,
                                               const void* gbase,
                                               unsigned gbyte_off) {
    asm volatile("global_load_async_to_lds_b128 %0, %1, %2"
                 :: "v"(lds_byte_addr), "v"(gbyte_off), "s"(gbase)
                 : "memory");
}

__global__ __launch_bounds__(256) void
mtp_p4_kernel(const float* __restrict__ x,
              const float* __restrict__ kern,    // [3,3,16,32] additive
              const float* __restrict__ tkern,   // [3,3,16,32] multiplicative
              float* __restrict__ out)           // [B,4,H,W,F]
{
    __shared__ float sK[9 * C_ * F_];   // 4608 floats
    __shared__ float sT[9 * C_ * F_];   // 4608 floats
    __shared__ float sX[HALO_H * HALO_W * C_];  // 1600 floats

    const int tid = threadIdx.x;
    const int x0  = blockIdx.x * TW;
    const int y0  = blockIdx.y * TH;
    const int b   = blockIdx.z;

    // ---- async-stage both structuring-element tensors into LDS (always in-bounds) ----
    #pragma unroll
    for (int i = 0; i < 5; ++i) {
        int idx = tid + i * 256;                 // float4 index
        if (idx < KVEC4) {
            unsigned goff  = (unsigned)(idx * 16);
            unsigned laddK = (unsigned)(uintptr_t)&sK[idx * 4];
            unsigned laddT = (unsigned)(uintptr_t)&sT[idx * 4];
            async_g2l_b128(laddK, kern,  goff);
            async_g2l_b128(laddT, tkern, goff);
        }
    }

    // ---- stage x halo tile (zero fill outside the image == TF SAME padding) ----
    #pragma unroll
    for (int i = 0; i < 2; ++i) {
        int idx = tid + i * 256;                 // float4 index over halo tile
        if (idx < XVEC4) {
            int c4   = idx & 3;                  // channel quartet
            int cell = idx >> 2;                 // 0..99
            int xx   = cell % HALO_W;
            int yy   = cell / HALO_W;
            int gy = y0 - 1 + yy, gx = x0 - 1 + xx;
            float4 v = make_float4(0.f, 0.f, 0.f, 0.f);
            if (gy >= 0 && gy < H_ && gx >= 0 && gx < W_) {
                v = *(const float4*)&x[(((size_t)b * H_ + gy) * W_ + gx) * C_ + c4 * 4];
            }
            *(float4*)&sX[(yy * HALO_W + xx) * C_ + c4 * 4] = v;
        }
    }

    asm volatile("s_wait_asynccnt 0" ::: "memory");   // async copies landed in LDS
    __syncthreads();                                   // make them visible block-wide

    const int lane = tid & 31;   // f  (F == wave32 width)
    const int py   = tid >> 5;   // pixel row in tile (8 waves)

    // P4 rotation tap permutations (np.rot90 of a 3x3, flattened row-major)
    constexpr int RM[4][9] = {
        {0, 1, 2, 3, 4, 5, 6, 7, 8},
        {2, 5, 8, 1, 4, 7, 0, 3, 6},
        {8, 7, 6, 5, 4, 3, 2, 1, 0},
        {6, 3, 0, 7, 4, 1, 8, 5, 2}};

    float acc[TW][4];
    #pragma unroll
    for (int px = 0; px < TW; ++px)
        #pragma unroll
        for (int r = 0; r < 4; ++r) acc[px][r] = 0.f;

    #pragma unroll 1
    for (int c = 0; c < C_; ++c) {
        // 9 taps of each SE for this channel, per-lane f slice -> registers
        float tk[9], kk[9];
        #pragma unroll
        for (int q = 0; q < 9; ++q) {
            tk[q] = sT[(q * C_ + c) * F_ + lane];   // conflict-free: lane-stride 4B
            kk[q] = sK[(q * C_ + c) * F_ + lane];
        }
        // 3 halo rows x 10 columns of x for this channel (LDS broadcast reads)
        float xv[3][HALO_W];
        #pragma unroll
        for (int row = 0; row < 3; ++row)
            #pragma unroll
            for (int xx = 0; xx < HALO_W; ++xx)
                xv[row][xx] = sX[((py + row) * HALO_W + xx) * C_ + c];

        #pragma unroll
        for (int px = 0; px < TW; ++px) {
            #pragma unroll
            for (int r = 0; r < 4; ++r) {
                float m = __builtin_fmaf(xv[0][px + 0], tk[RM[r][0]], kk[RM[r][0]]);
                #pragma unroll
                for (int p = 1; p < 9; ++p) {
                    const int ki = p / 3, kj = p % 3;
                    m = fmaxf(m, __builtin_fmaf(xv[ki][px + kj],
                                                tk[RM[r][p]], kk[RM[r][p]]));
                }
                acc[px][r] += m;   // tropical max over taps, then plain sum over c
            }
        }
    }

    // ---- streamed stores: out[b, r, y, x, f], one 128B transaction per store ----
    const int gy = y0 + py;
    #pragma unroll
    for (int px = 0; px < TW; ++px) {
        const int gx = x0 + px;
        #pragma unroll
        for (int r = 0; r < 4; ++r) {
            float* p = &out[((((size_t)b * 4 + r) * H_ + gy) * W_ + gx) * F_ + lane];
            __builtin_nontemporal_store(acc[px][r], p);
        }
    }
}

extern "C" void kernel_launch(void* const* d_in, const int* in_sizes, int n_in,
                              void* d_out, int out_size, void* d_ws, size_t ws_size,
                              hipStream_t stream) {
    const float* x     = (const float*)d_in[0];
    const float* kern  = (const float*)d_in[1];
    const float* tkern = (const float*)d_in[2];
    float* out = (float*)d_out;

    dim3 grid(W_ / TW, H_ / TH, B_);   // (8, 8, 4) = 256 blocks
    dim3 block(256);                    // 8 wave32 waves; lane = output feature
    mtp_p4_kernel<<<grid, block, 0, stream>>>(x, kern, tkern, out);
}